// GCN_26697516712651
// MI455X (gfx1250) — compile-verified
//
#include <hip/hip_runtime.h>
#include <hip/hip_bf16.h>

typedef float v2f __attribute__((ext_vector_type(2)));
typedef float v8f __attribute__((ext_vector_type(8)));

#define NFEAT 128

// ---------------------------------------------------------------- deg / dinv
__global__ void gcn_fill1(float* __restrict__ deg, int n) {
    int i = blockIdx.x * blockDim.x + threadIdx.x;
    if (i < n) deg[i] = 1.0f;                       // self-loop weight
}

__global__ void gcn_deg_accum(const int* __restrict__ dst,
                              const float* __restrict__ ew,
                              float* __restrict__ deg, int n_edges) {
    int e = blockIdx.x * blockDim.x + threadIdx.x;
    if (e < n_edges)
        __hip_atomic_fetch_add(&deg[dst[e]], ew[e],
                               __ATOMIC_RELAXED, __HIP_MEMORY_SCOPE_AGENT);
}

__global__ void gcn_rsqrt(float* __restrict__ d, int n) {
    int i = blockIdx.x * blockDim.x + threadIdx.x;
    if (i < n) d[i] = rsqrtf(d[i]);
}

__global__ void gcn_zero(float* __restrict__ p, int n) {
    int i = blockIdx.x * blockDim.x + threadIdx.x;
    if (i < n) p[i] = 0.0f;
}

// ---------------------------------------------------------------- WMMA GEMM
// H[M,128] = X[M,128] @ W[128,128], f32 exact via V_WMMA_F32_16X16X4_F32.
// block = (32,8): 8 waves, each wave owns one 16x16 N-tile of the block's M-tile.
__global__ __launch_bounds__(256) void gcn_gemm128_wmma(
        const float* __restrict__ X, const float* __restrict__ W,
        float* __restrict__ H) {
    __shared__ float Wl[NFEAT * NFEAT];             // 64 KB of 320 KB LDS
    {
        const float4* Wv  = (const float4*)W;
        float4*       Wlv = (float4*)Wl;
        int tid = threadIdx.y * 32 + threadIdx.x;
        #pragma unroll
        for (int i = 0; i < (NFEAT * NFEAT / 4) / 256; ++i)
            Wlv[tid + i * 256] = Wv[tid + i * 256];
    }
    __syncthreads();

    const int lane  = threadIdx.x;                  // 0..31
    const int m0    = blockIdx.x * 16;
    const int n0    = threadIdx.y * 16;
    const int mlane = lane & 15;
    const int khalf = (lane >> 4) * 2;              // 0 or 2 (A/B K placement)
    const int ncol  = n0 + mlane;

    const float* xrow = X + (size_t)(m0 + mlane) * NFEAT;
    v8f acc = {};

    #pragma unroll
    for (int k0 = 0; k0 < NFEAT; k0 += 4) {
        const int k = k0 + khalf;
        v2f a = *(const v2f*)&xrow[k];              // K=k, k+1 of row m
        v2f b;
        b.x = Wl[(k    ) * NFEAT + ncol];           // B row K=k,   col n
        b.y = Wl[(k + 1) * NFEAT + ncol];           // B row K=k+1, col n
        acc = __builtin_amdgcn_wmma_f32_16x16x4_f32(
                  false, a, false, b, (short)0, acc, false, false);
    }

    const int rbase = m0 + (lane >> 4) * 8;         // D layout: M = r + 8*(lane>=16)
    #pragma unroll
    for (int r = 0; r < 8; ++r)
        H[(size_t)(rbase + r) * NFEAT + ncol] = acc[r];
}

// ---------------------------------------------------------------- edge scatter
// One wave32 per edge; each lane moves 4 consecutive floats (float4 gather +
// 4 global f32 atomic adds).
__global__ __launch_bounds__(256) void gcn_scatter(
        const float* __restrict__ h, float* __restrict__ agg,
        const int* __restrict__ src, const int* __restrict__ dst,
        const float* __restrict__ ew, const float* __restrict__ dinv,
        int n_edges) {
    const int wave = (blockIdx.x * blockDim.x + threadIdx.x) >> 5;
    const int lane = threadIdx.x & 31;
    if (wave >= n_edges) return;

    const int   s = src[wave];
    const int   d = dst[wave];
    const float c = dinv[s] * ew[wave] * dinv[d];

    const float4 hv = *(const float4*)&h[(size_t)s * NFEAT + lane * 4];
    float* out = &agg[(size_t)d * NFEAT + lane * 4];
    __hip_atomic_fetch_add(&out[0], hv.x * c, __ATOMIC_RELAXED, __HIP_MEMORY_SCOPE_AGENT);
    __hip_atomic_fetch_add(&out[1], hv.y * c, __ATOMIC_RELAXED, __HIP_MEMORY_SCOPE_AGENT);
    __hip_atomic_fetch_add(&out[2], hv.z * c, __ATOMIC_RELAXED, __HIP_MEMORY_SCOPE_AGENT);
    __hip_atomic_fetch_add(&out[3], hv.w * c, __ATOMIC_RELAXED, __HIP_MEMORY_SCOPE_AGENT);
}

// ---------------------------------------------------------------- self-loop + bias (+ReLU)
// out[i,d] = act( agg[i,d] + hpre[i,d]*dinv[i]^2 + b[d] )
__global__ void gcn_fuse(const float* __restrict__ agg,
                         const float* __restrict__ hpre,
                         const float* __restrict__ dinv,
                         const float* __restrict__ b,
                         float* __restrict__ out,
                         int n_nodes, int do_relu) {
    int idx = blockIdx.x * blockDim.x + threadIdx.x;
    if (idx >= n_nodes * NFEAT) return;
    int i = idx >> 7;
    int d = idx & (NFEAT - 1);
    float di = dinv[i];
    float v = agg[idx] + hpre[idx] * (di * di) + b[d];
    out[idx] = do_relu ? fmaxf(v, 0.0f) : v;
}

// ---------------------------------------------------------------- classifier
__global__ void gcn_classifier(const float* __restrict__ h,
                               const float* __restrict__ Wc,
                               const float* __restrict__ bc,
                               float* __restrict__ out, int n_nodes) {
    int i = blockIdx.x * blockDim.x + threadIdx.x;
    if (i >= n_nodes) return;
    float acc[10];
    #pragma unroll
    for (int c = 0; c < 10; ++c) acc[c] = bc[c];
    const float* hr = h + (size_t)i * NFEAT;
    for (int d = 0; d < NFEAT; ++d) {
        float hv = hr[d];
        #pragma unroll
        for (int c = 0; c < 10; ++c) acc[c] += hv * Wc[d * 10 + c];
    }
    #pragma unroll
    for (int c = 0; c < 10; ++c) out[(size_t)i * 10 + c] = acc[c];
}

// ---------------------------------------------------------------- launcher
extern "C" void kernel_launch(void* const* d_in, const int* in_sizes, int n_in,
                              void* d_out, int out_size, void* d_ws, size_t ws_size,
                              hipStream_t stream) {
    const float* x   = (const float*)d_in[0];
    const int*   ei  = (const int*)  d_in[1];
    const float* ew  = (const float*)d_in[2];
    const float* W1  = (const float*)d_in[3];
    const float* b1  = (const float*)d_in[4];
    const float* W2  = (const float*)d_in[5];
    const float* b2  = (const float*)d_in[6];
    const float* Wc  = (const float*)d_in[7];
    const float* bc  = (const float*)d_in[8];
    float*       out = (float*)d_out;

    const int n_nodes = in_sizes[0] / NFEAT;        // 50000
    const int n_edges = in_sizes[2];                // 600000
    const int* src = ei;
    const int* dst = ei + n_edges;

    const size_t nd = (size_t)n_nodes * NFEAT;      // 6.4M floats
    float* bufA = (float*)d_ws;                     // h-pre / h1
    float* bufB = bufA + nd;                        // agg1 / h2-pre / h2
    float* bufC = bufB + nd;                        // agg2
    float* dinv = bufC + nd;                        // deg -> dinv

    const int B = 256;
    const int gN  = (n_nodes + B - 1) / B;
    const int gE  = (n_edges + B - 1) / B;
    const int gND = (int)((nd + B - 1) / B);
    const int gEw = (int)(((size_t)n_edges * 32 + B - 1) / B);   // wave/edge
    const dim3 gblk(32, 8);
    const int gTiles = n_nodes / 16;                // 3125, exact

    // degrees -> dinv
    gcn_fill1    <<<gN, B, 0, stream>>>(dinv, n_nodes);
    gcn_deg_accum<<<gE, B, 0, stream>>>(dst, ew, dinv, n_edges);
    gcn_rsqrt    <<<gN, B, 0, stream>>>(dinv, n_nodes);

    // layer 1
    gcn_gemm128_wmma<<<gTiles, gblk, 0, stream>>>(x, W1, bufA);
    gcn_zero        <<<gND, B, 0, stream>>>(bufB, (int)nd);
    gcn_scatter     <<<gEw, B, 0, stream>>>(bufA, bufB, src, dst, ew, dinv, n_edges);
    gcn_fuse        <<<gND, B, 0, stream>>>(bufB, bufA, dinv, b1, bufA, n_nodes, 1);

    // layer 2
    gcn_gemm128_wmma<<<gTiles, gblk, 0, stream>>>(bufA, W2, bufB);
    gcn_zero        <<<gND, B, 0, stream>>>(bufC, (int)nd);
    gcn_scatter     <<<gEw, B, 0, stream>>>(bufB, bufC, src, dst, ew, dinv, n_edges);
    gcn_fuse        <<<gND, B, 0, stream>>>(bufC, bufB, dinv, b2, bufB, n_nodes, 1);

    // classifier
    gcn_classifier<<<gN, B, 0, stream>>>(bufB, Wc, bc, out, n_nodes);
}